// GCN_80942953661103
// MI455X (gfx1250) — compile-verified
//
#include <hip/hip_runtime.h>

// ---------------------------------------------------------------------------
// GCN 2-layer forward for MI455X (gfx1250, wave32).
// GEMMs use V_WMMA_F32_16X16X4_F32 (full fp32 precision, matches reference).
// Edge aggregation is the memory-bound part: one wave32 per edge, hardware
// float atomics (global_atomic_add_f32) for the scatter-add.
// ---------------------------------------------------------------------------

typedef float v2f __attribute__((ext_vector_type(2)));
typedef float v8f __attribute__((ext_vector_type(8)));

#define D_IN  96
#define D_HID 96
#define D_OUT 48

// ---------------- degree / normalization ----------------

__global__ void deg_kernel(const int* __restrict__ dst, float* __restrict__ deg, int E) {
    int e = blockIdx.x * blockDim.x + threadIdx.x;
    if (e < E) unsafeAtomicAdd(&deg[dst[e]], 1.0f);
}

__global__ void dinv_kernel(float* __restrict__ deg, int N) {
    int i = blockIdx.x * blockDim.x + threadIdx.x;
    if (i < N) {
        float d = deg[i] + 1.0f;           // +1 self-loop
        deg[i] = __frsqrt_rn(d);           // deg>0 always
    }
}

// ---------------- fp32 WMMA GEMM: C[M,N] = A[M,K] * B[K,N] ----------------
// Requires M%16==0, N%16==0, K%4==0 (true here: M=50000, N in {96,48}, K=96).
// Per-wave 16x16 output tile, K-loop in steps of 4 via v_wmma_f32_16x16x4_f32.

__global__ void gemm_wmma_f32(const float* __restrict__ A,
                              const float* __restrict__ B,
                              float* __restrict__ C,
                              int M, int N, int K) {
    const int lane  = threadIdx.x & 31;
    const int wave  = threadIdx.x >> 5;
    const int wpb   = blockDim.x >> 5;
    const int tile  = blockIdx.x * wpb + wave;
    const int tilesN = N >> 4;
    const int tilesM = M >> 4;
    if (tile >= tilesM * tilesN) return;   // wave-uniform: EXEC stays all-ones

    const int tm = tile / tilesN;
    const int tn = tile - tm * tilesN;
    const int m0 = tm << 4;
    const int n0 = tn << 4;

    const int lm = lane & 15;              // row (A) / col (B,C) within tile
    const int hi = lane >> 4;              // lane-half selector

    const float* __restrict__ arow = A + (size_t)(m0 + lm) * K;
    const float* __restrict__ bcol = B + n0 + lm;

    v8f acc = {};
    for (int k0 = 0; k0 < K; k0 += 4) {
        const int ka = k0 + 2 * hi;        // per-lane K pair: ka, ka+1
        v2f a, b;
        a.x = arow[ka];
        a.y = arow[ka + 1];
        b.x = bcol[(size_t)ka * N];
        b.y = bcol[(size_t)(ka + 1) * N];
        // 8 args: (neg_a, A, neg_b, B, c_mod, C, reuse_a, reuse_b)
        acc = __builtin_amdgcn_wmma_f32_16x16x4_f32(
            false, a, false, b, (short)0, acc, false, false);
    }

    // C/D layout: VGPR r holds row (m0 + r + 8*hi), col (n0 + lm)
    float* __restrict__ crow = C + (size_t)(m0 + 8 * hi) * N + n0 + lm;
#pragma unroll
    for (int r = 0; r < 8; ++r)
        crow[(size_t)r * N] = acc[r];
}

// ---------------- edge scatter: out[dst] += dinv[s]*dinv[d] * h[src] --------
// One wave32 per edge; lane strides the feature dimension.

template <int F>
__global__ void scatter_edges(const float* __restrict__ h,
                              const int* __restrict__ src,
                              const int* __restrict__ dst,
                              const float* __restrict__ dinv,
                              float* __restrict__ out, int E) {
    const int wave = (int)((blockIdx.x * (size_t)blockDim.x + threadIdx.x) >> 5);
    const int lane = threadIdx.x & 31;
    if (wave >= E) return;
    const int s = src[wave];
    const int d = dst[wave];
    const float norm = dinv[s] * dinv[d];
    const float* __restrict__ hs = h + (size_t)s * F;
    float* __restrict__ od = out + (size_t)d * F;
#pragma unroll
    for (int c = lane; c < F; c += 32)
        unsafeAtomicAdd(&od[c], hs[c] * norm);
}

// ---------------- self-loop + bias (+ReLU) fused epilogue ------------------
// out[i][c] = act( out[i][c] + dinv[i]^2 * h[i][c] + bias[c] )

template <int F, bool RELU>
__global__ void epilogue(const float* __restrict__ h,
                         const float* __restrict__ dinv,
                         const float* __restrict__ bias,
                         float* __restrict__ out, int N) {
    const size_t idx = blockIdx.x * (size_t)blockDim.x + threadIdx.x;
    const size_t total = (size_t)N * F;
    if (idx >= total) return;
    const int node = (int)(idx / F);
    const int c    = (int)(idx - (size_t)node * F);
    const float di = dinv[node];
    float v = out[idx] + di * di * h[idx] + bias[c];
    if (RELU) v = fmaxf(v, 0.0f);
    out[idx] = v;
}

// ---------------------------------------------------------------------------

extern "C" void kernel_launch(void* const* d_in, const int* in_sizes, int n_in,
                              void* d_out, int out_size, void* d_ws, size_t ws_size,
                              hipStream_t stream) {
    const float* x   = (const float*)d_in[0];
    const int*   ei  = (const int*)d_in[1];   // [2, E] (JAX x64 off -> int32)
    const float* W1  = (const float*)d_in[2];
    const float* b1  = (const float*)d_in[3];
    const float* W2  = (const float*)d_in[4];
    const float* b2  = (const float*)d_in[5];
    float* out = (float*)d_out;

    const int Nn = in_sizes[0] / D_IN;        // 50000
    const int E  = in_sizes[1] / 2;           // 800000
    const int* src = ei;
    const int* dst = ei + E;

    // workspace carve-up (bytes)
    char* ws = (char*)d_ws;
    const size_t OFF_DINV = 0;                               // Nn f32
    const size_t OFF_H1   = 204800;                          // Nn*96 f32
    const size_t OFF_AGG1 = OFF_H1 + (size_t)Nn * D_HID * 4; // Nn*96 f32
    const size_t OFF_H2   = OFF_AGG1 + (size_t)Nn * D_HID * 4; // Nn*48 f32
    float* dinv = (float*)(ws + OFF_DINV);
    float* h1   = (float*)(ws + OFF_H1);
    float* agg1 = (float*)(ws + OFF_AGG1);
    float* h2   = (float*)(ws + OFF_H2);

    // zero accumulators (capture-safe)
    hipMemsetAsync(dinv, 0, (size_t)Nn * 4, stream);
    hipMemsetAsync(agg1, 0, (size_t)Nn * D_HID * 4, stream);
    hipMemsetAsync(out,  0, (size_t)Nn * D_OUT * 4, stream);

    // degree -> dinv
    deg_kernel<<<(E + 255) / 256, 256, 0, stream>>>(dst, dinv, E);
    dinv_kernel<<<(Nn + 255) / 256, 256, 0, stream>>>(dinv, Nn);

    // layer 1: h1 = x @ W1
    {
        const int tiles = (Nn / 16) * (D_HID / 16);
        const int blocks = (tiles + 7) / 8;            // 8 waves / block
        gemm_wmma_f32<<<blocks, 256, 0, stream>>>(x, W1, h1, Nn, D_HID, D_IN);
    }
    scatter_edges<D_HID><<<(int)(((size_t)E * 32 + 255) / 256), 256, 0, stream>>>(
        h1, src, dst, dinv, agg1, E);
    epilogue<D_HID, true><<<(int)(((size_t)Nn * D_HID + 255) / 256), 256, 0, stream>>>(
        h1, dinv, b1, agg1, Nn);

    // layer 2: h2 = act1 @ W2
    {
        const int tiles = (Nn / 16) * (D_OUT / 16);
        const int blocks = (tiles + 7) / 8;
        gemm_wmma_f32<<<blocks, 256, 0, stream>>>(agg1, W2, h2, Nn, D_OUT, D_HID);
    }
    scatter_edges<D_OUT><<<(int)(((size_t)E * 32 + 255) / 256), 256, 0, stream>>>(
        h2, src, dst, dinv, out, E);
    epilogue<D_OUT, false><<<(int)(((size_t)Nn * D_OUT + 255) / 256), 256, 0, stream>>>(
        h2, dinv, b2, out, Nn);
}